// HeAttention_4123168604567
// MI455X (gfx1250) — compile-verified
//
#include <hip/hip_runtime.h>
#include <hip/hip_bf16.h>

typedef float v2f __attribute__((ext_vector_type(2)));
typedef float v8f __attribute__((ext_vector_type(8)));

#define S_LEN 512
#define D_MODEL 512
#define N_HEADS 8
#define HEAD_DIM 64
#define N_EMB 1024

// ---------------------------------------------------------------------------
// Wave-level f32 WMMA GEMM, 16x64 C-strip per wave (4 independent
// accumulator chains), 3-stage software-pipelined K loop (prefetch dist 2).
//   BT = true :  C = A * B^T (+bias)   A[M,K] lda, B[N,K] ldb  (torch Linear)
//   BT = false:  C = A * B   (+bias)   A[M,K] lda, B[K,N] ldb
// amdgpu_waves_per_eu(1) lifts the default 64-VGPR occupancy cap (rounds 2/3
// showed the allocator recycling one B register pair -> s_wait_loadcnt 0x0
// before every WMMA). Two load groups (10 b64) stay in flight in distinct
// registers; cross-wave parallelism comes from the 256-512 block grid.
//
// f32 WMMA VGPR layout (ISA 7.12.2):
//   A 16x4 : lane l -> row m = l&15 ; VGPR v, half = l>=16 -> k = v + 2*half
//   B 4x16 : lane l -> col n = l&15 ; VGPR v, half        -> k = v + 2*half
//   C 16x16: lane l -> col n = l&15 ; VGPR v              -> m = v + 8*half
// ---------------------------------------------------------------------------
template <bool BT>
__device__ __forceinline__ void load_b4(const float* __restrict__ B, int ldb,
                                        int n0, int r, int kk, int k0,
                                        v2f b[4]) {
  if (BT) {
    const float* bp = B + (size_t)(n0 + r) * ldb + k0 + kk;
    b[0].x = bp[0];                  b[0].y = bp[1];
    b[1].x = bp[(size_t)16 * ldb];   b[1].y = bp[(size_t)16 * ldb + 1];
    b[2].x = bp[(size_t)32 * ldb];   b[2].y = bp[(size_t)32 * ldb + 1];
    b[3].x = bp[(size_t)48 * ldb];   b[3].y = bp[(size_t)48 * ldb + 1];
  } else {
    const float* bp = B + (size_t)(k0 + kk) * ldb + (n0 + r);
    b[0].x = bp[0];   b[0].y = bp[ldb];
    b[1].x = bp[16];  b[1].y = bp[ldb + 16];
    b[2].x = bp[32];  b[2].y = bp[ldb + 32];
    b[3].x = bp[48];  b[3].y = bp[ldb + 48];
  }
}

#define WMMA4(a, b, c0, c1, c2, c3)                                           \
  do {                                                                        \
    c0 = __builtin_amdgcn_wmma_f32_16x16x4_f32(false, (a), false, (b)[0],     \
                                               (short)0, c0, false, false);   \
    c1 = __builtin_amdgcn_wmma_f32_16x16x4_f32(false, (a), false, (b)[1],     \
                                               (short)0, c1, false, false);   \
    c2 = __builtin_amdgcn_wmma_f32_16x16x4_f32(false, (a), false, (b)[2],     \
                                               (short)0, c2, false, false);   \
    c3 = __builtin_amdgcn_wmma_f32_16x16x4_f32(false, (a), false, (b)[3],     \
                                               (short)0, c3, false, false);   \
  } while (0)

template <bool BT>
__global__ __launch_bounds__(32)
__attribute__((amdgpu_waves_per_eu(1))) void wmma_gemm_f32(
    const float* __restrict__ A, const float* __restrict__ B,
    const float* __restrict__ bias, float* __restrict__ C,
    int M, int N, int Kd, int lda, int ldb, int ldc,
    long long aHead, long long bHead, long long cHead) {
  const int h = blockIdx.z;
  A += (long long)h * aHead;
  B += (long long)h * bHead;
  C += (long long)h * cHead;

  const int n0 = blockIdx.x * 64;   // 4 x 16 columns per wave
  const int m0 = blockIdx.y * 16;
  const int lane = threadIdx.x;     // blockDim.x == 32 (one wave)
  const int r    = lane & 15;       // row for A, col for B/C
  const int half = lane >> 4;       // 0 or 1
  const int kk   = half * 2;        // K sub-offset within a 4-step

  v8f acc0 = {0.f, 0.f, 0.f, 0.f, 0.f, 0.f, 0.f, 0.f};
  v8f acc1 = acc0, acc2 = acc0, acc3 = acc0;

  const float* ap = A + (size_t)(m0 + r) * lda + kk;

  // ---- 3-stage software pipeline: stage steps 0 and 1 ----
  v2f aS0, aS1, bS0[4], bS1[4];
  aS0.x = ap[0];
  aS0.y = ap[1];
  load_b4<BT>(B, ldb, n0, r, kk, 0, bS0);
  aS1.x = ap[4];
  aS1.y = ap[5];
  load_b4<BT>(B, ldb, n0, r, kk, 4, bS1);

  const int ns = Kd >> 2;  // number of K-steps (>= 2 for all call sites)
  for (int i = 0; i < ns - 2; ++i) {
    // issue loads for step i+2 into fresh temporaries
    v2f aS2, bS2[4];
    const int kN = (i + 2) * 4;
    aS2.x = ap[kN];
    aS2.y = ap[kN + 1];
    load_b4<BT>(B, ldb, n0, r, kk, kN, bS2);

    // consume step i
    WMMA4(aS0, bS0, acc0, acc1, acc2, acc3);

    // rotate
    aS0 = aS1;
    bS0[0] = bS1[0]; bS0[1] = bS1[1]; bS0[2] = bS1[2]; bS0[3] = bS1[3];
    aS1 = aS2;
    bS1[0] = bS2[0]; bS1[1] = bS2[1]; bS1[2] = bS2[2]; bS1[3] = bS2[3];
  }

  // ---- epilogue: last two steps ----
  WMMA4(aS0, bS0, acc0, acc1, acc2, acc3);
  WMMA4(aS1, bS1, acc0, acc1, acc2, acc3);

  const int mbase = m0 + half * 8;
  float bv0 = 0.f, bv1 = 0.f, bv2 = 0.f, bv3 = 0.f;
  if (bias) {
    bv0 = bias[n0 + r];
    bv1 = bias[n0 + 16 + r];
    bv2 = bias[n0 + 32 + r];
    bv3 = bias[n0 + 48 + r];
  }
#pragma unroll
  for (int v = 0; v < 8; ++v) {
    float* cr = C + (size_t)(mbase + v) * ldc + (n0 + r);
    cr[0]  = acc0[v] + bv0;
    cr[16] = acc1[v] + bv1;
    cr[32] = acc2[v] + bv2;
    cr[48] = acc3[v] + bv3;
  }
}

// ---------------------------------------------------------------------------
// scores[h,i,j] = (term1[h,i,j] + T2[h, i, j-i+512] + T3[h, j-i+512, j]) / 8
// then row-softmax over j; result written straight into attn slot of d_out.
// One block (256 threads) per (i, h) row; 2 elements per thread.
// ---------------------------------------------------------------------------
__global__ void combine_softmax(const float* __restrict__ s1,
                                const float* __restrict__ T2,
                                const float* __restrict__ T3,
                                float* __restrict__ attn) {
  const int i = blockIdx.x;
  const int h = blockIdx.y;
  const int t = threadIdx.x;  // 256 threads

  const float* s1r = s1 + ((size_t)h * S_LEN + i) * S_LEN;
  const float* t2r = T2 + (size_t)h * S_LEN * N_EMB + (size_t)i * N_EMB;
  const float* t3h = T3 + (size_t)h * N_EMB * S_LEN;
  float* ar = attn + ((size_t)h * S_LEN + i) * S_LEN;

  const float scale = 0.125f;  // 1/sqrt(64)

  int j0 = t;
  int p0 = j0 - i + 512;
  float v0 = (s1r[j0] + t2r[p0] + t3h[(size_t)p0 * S_LEN + j0]) * scale;
  int j1 = t + 256;
  int p1 = j1 - i + 512;
  float v1 = (s1r[j1] + t2r[p1] + t3h[(size_t)p1 * S_LEN + j1]) * scale;

  __shared__ float red[256];

  // max reduce
  red[t] = fmaxf(v0, v1);
  __syncthreads();
  for (int s = 128; s > 0; s >>= 1) {
    if (t < s) red[t] = fmaxf(red[t], red[t + s]);
    __syncthreads();
  }
  const float m = red[0];
  __syncthreads();

  // exp + sum reduce
  const float e0 = __expf(v0 - m);
  const float e1 = __expf(v1 - m);
  red[t] = e0 + e1;
  __syncthreads();
  for (int s = 128; s > 0; s >>= 1) {
    if (t < s) red[t] += red[t + s];
    __syncthreads();
  }
  const float inv = 1.0f / red[0];

  ar[j0] = e0 * inv;
  ar[j1] = e1 * inv;
}

extern "C" void kernel_launch(void* const* d_in, const int* in_sizes, int n_in,
                              void* d_out, int out_size, void* d_ws, size_t ws_size,
                              hipStream_t stream) {
  const float* hid = (const float*)d_in[0];   // [512, 512]
  const float* emb = (const float*)d_in[1];   // [1024, 512]
  const float* wq  = (const float*)d_in[2];
  const float* bq  = (const float*)d_in[3];
  const float* wk  = (const float*)d_in[4];
  const float* bk  = (const float*)d_in[5];
  const float* wv  = (const float*)d_in[6];
  const float* bv  = (const float*)d_in[7];
  const float* wrk = (const float*)d_in[8];
  const float* brk = (const float*)d_in[9];
  const float* wrq = (const float*)d_in[10];
  const float* brq = (const float*)d_in[11];
  const float* wo  = (const float*)d_in[12];
  const float* bo  = (const float*)d_in[13];

  float* out  = (float*)d_out;                            // [512, 512]
  float* attn = (float*)d_out + (size_t)S_LEN * D_MODEL;  // [8, 512, 512]

  // workspace layout (floats), ~50 MB total
  float* ws  = (float*)d_ws;
  float* Q   = ws;                       // 512*512
  float* K   = Q + 262144;               // 512*512
  float* V   = K + 262144;               // 512*512
  float* RK  = V + 262144;               // 1024*512
  float* RQ  = RK + 524288;              // 1024*512
  float* s1  = RQ + 524288;              // 8*512*512
  float* T2  = s1 + 2097152;             // 8*512*1024
  float* T3  = T2 + 4194304;             // 8*1024*512
  float* ctx = T3 + 4194304;             // 512*512

  const dim3 blk(32, 1, 1);

  // --- projections: C = A*W^T + b ---  (grid.x covers N/64)
  wmma_gemm_f32<true><<<dim3(8, 32, 1), blk, 0, stream>>>(
      hid, wq, bq, Q, 512, 512, 512, 512, 512, 512, 0, 0, 0);
  wmma_gemm_f32<true><<<dim3(8, 32, 1), blk, 0, stream>>>(
      hid, wk, bk, K, 512, 512, 512, 512, 512, 512, 0, 0, 0);
  wmma_gemm_f32<true><<<dim3(8, 32, 1), blk, 0, stream>>>(
      hid, wv, bv, V, 512, 512, 512, 512, 512, 512, 0, 0, 0);
  wmma_gemm_f32<true><<<dim3(8, 64, 1), blk, 0, stream>>>(
      emb, wrk, brk, RK, 1024, 512, 512, 512, 512, 512, 0, 0, 0);
  wmma_gemm_f32<true><<<dim3(8, 64, 1), blk, 0, stream>>>(
      emb, wrq, brq, RQ, 1024, 512, 512, 512, 512, 512, 0, 0, 0);

  // --- per-head score GEMMs (grid.z = head) ---
  // term1[h] = Q_h * K_h^T           [512x64]x[64x512]
  wmma_gemm_f32<true><<<dim3(8, 32, N_HEADS), blk, 0, stream>>>(
      Q, K, nullptr, s1, 512, 512, HEAD_DIM, 512, 512, 512,
      HEAD_DIM, HEAD_DIM, (long long)512 * 512);
  // T2[h][i][p] = Q_h * RK_h^T       [512x64]x[64x1024]
  wmma_gemm_f32<true><<<dim3(16, 32, N_HEADS), blk, 0, stream>>>(
      Q, RK, nullptr, T2, 512, 1024, HEAD_DIM, 512, 512, 1024,
      HEAD_DIM, HEAD_DIM, (long long)512 * 1024);
  // T3[h][p][j] = RQ_h * K_h^T       [1024x64]x[64x512]
  wmma_gemm_f32<true><<<dim3(8, 64, N_HEADS), blk, 0, stream>>>(
      RQ, K, nullptr, T3, 1024, 512, HEAD_DIM, 512, 512, 512,
      HEAD_DIM, HEAD_DIM, (long long)1024 * 512);

  // --- gather the Toeplitz terms + scaled softmax -> attn (in d_out) ---
  combine_softmax<<<dim3(S_LEN, N_HEADS, 1), dim3(256, 1, 1), 0, stream>>>(
      s1, T2, T3, attn);

  // --- ctx_h = attn_h * V_h         [512x512]x[512x64], merged heads in ctx ---
  wmma_gemm_f32<false><<<dim3(1, 32, N_HEADS), blk, 0, stream>>>(
      attn, V, nullptr, ctx, 512, HEAD_DIM, 512, 512, 512, 512,
      (long long)512 * 512, HEAD_DIM, HEAD_DIM);

  // --- out = ctx * Wo^T + bo ---
  wmma_gemm_f32<true><<<dim3(8, 32, 1), blk, 0, stream>>>(
      ctx, wo, bo, out, 512, 512, 512, 512, 512, 512, 0, 0, 0);
}